// VertexDistExtract3D_65360812311305
// MI455X (gfx1250) — compile-verified
//
#include <hip/hip_runtime.h>

typedef float v2f __attribute__((ext_vector_type(2)));
typedef float v8f __attribute__((ext_vector_type(8)));

#define D_IN  199   // 7 + 3*64
#define K_V   64
#define P_D   4

// Branch-free 4-way select: lowers to 3 v_cndmask (no exec-mask diamonds,
// no scratch-backed dynamic indexing).
__device__ __forceinline__ float sel4(int j, float a0, float a1, float a2, float a3) {
  const float t01 = (j & 1) ? a1 : a0;
  const float t23 = (j & 1) ? a3 : a2;
  return (j & 2) ? t23 : t01;
}

// One wave32 = one point. Bandwidth-bound (~238 MB traffic -> ~10us at
// 23.3 TB/s). out[k,p] = Mt[p]·v_k + c[p] with Mt = dirs@R, c = dirs@pos,
// evaluated as four V_WMMA_F32_16X16X4_F32 per point (full fp32 precision).
__global__ __launch_bounds__(256) void vdx_kernel(const float* __restrict__ x,
                                                  const float* __restrict__ dirs,
                                                  float* __restrict__ out,
                                                  int npts) {
  const int lane = threadIdx.x & 31;
  const int wave = threadIdx.x >> 5;
  const int pt   = blockIdx.x * 8 + wave;
  if (pt >= npts) return;  // wave-uniform: EXEC stays all-1 for WMMA

  const float* row = x + (size_t)pt * D_IN;

  // pos / quat: same address across lanes -> one cacheline per wave
  const float px = row[0], py = row[1], pz = row[2];
  const float qw = row[3], qx = row[4], qy = row[5], qz = row[6];

  // Rotation matrix from quaternion (unit formula, per reference)
  const float xx = qx*qx, yy = qy*qy, zz = qz*qz;
  const float xy = qx*qy, xz = qx*qz, yz = qy*qz;
  const float wx = qw*qx, wy = qw*qy, wz = qw*qz;
  const float R00 = 1.f - 2.f*(yy+zz), R01 = 2.f*(xy-wz),       R02 = 2.f*(xz+wy);
  const float R10 = 2.f*(xy+wz),       R11 = 1.f - 2.f*(xx+zz), R12 = 2.f*(yz-wx);
  const float R20 = 2.f*(xz-wy),       R21 = 2.f*(yz+wx),       R22 = 1.f - 2.f*(xx+yy);

  const int  pcol   = lane & 3;          // this lane's output direction p
  const int  jgrp   = (lane >> 2) & 3;   // replicated column group
  const int  half   = lane >> 4;         // row half (M 0-7 vs 8-15)
  const bool lohalf = (lane < 16);

  // dirs is uniform (s_load); pick this lane's direction via cndmask tree,
  // then compute ONLY this lane's Mt row + offset. No private arrays.
  const float d0 = sel4(pcol, dirs[0], dirs[3], dirs[6], dirs[9]);
  const float d1 = sel4(pcol, dirs[1], dirs[4], dirs[7], dirs[10]);
  const float d2 = sel4(pcol, dirs[2], dirs[5], dirs[8], dirs[11]);
  const float mt0 = d0*R00 + d1*R10 + d2*R20;   // Mt[p][0]
  const float mt1 = d0*R01 + d1*R11 + d2*R21;   // Mt[p][1]
  const float mt2 = d0*R02 + d1*R12 + d2*R22;   // Mt[p][2]
  const float cc  = d0*px  + d1*py  + d2*pz;    // c[p]

  // B fragment (4x16 f32): lanes 0-15 hold rows K=0,1; lanes 16-31 rows K=2,3.
  // Columns replicated mod 4 so every lane's D column is meaningful.
  v2f bfrag;
  bfrag.x = lohalf ? mt0 : mt2;
  bfrag.y = lohalf ? mt1 : cc;

  const float* verts = row + 7;
  float* orow = out + (size_t)pt * (K_V * P_D);
  const int m = lane & 15;

#pragma unroll
  for (int kc = 0; kc < K_V; kc += 16) {
    // A fragment (16x4 f32): lanes 0-15 -> K=0,1 = (vx,vy);
    // lanes 16-31 -> K=2,3 = (vz, 1.0)
    const float* v3 = verts + 3 * (kc + m);
    const float e0 = v3[lohalf ? 0 : 2];   // one load, selected address
    const float e1 = v3[1];                // unconditional load, select after
    v2f afrag;
    afrag.x = e0;
    afrag.y = lohalf ? e1 : 1.0f;

    v8f acc = {};
    acc = __builtin_amdgcn_wmma_f32_16x16x4_f32(false, afrag, false, bfrag,
                                                (short)0, acc, false, false);

    // D[r]: lanes<16 -> M=r, lanes>=16 -> M=8+r; column n == p (mod 4).
    // Lane (16*half + 4*j + p) stores rows 8*half+j and 8*half+4+j, so each
    // global_store_b32 covers two full contiguous 64B cachelines per wave.
    const float s0 = sel4(jgrp, acc[0], acc[1], acc[2], acc[3]);
    const float s1 = sel4(jgrp, acc[4], acc[5], acc[6], acc[7]);
    orow[(kc + half*8 + jgrp)     * 4 + pcol] = s0;
    orow[(kc + half*8 + 4 + jgrp) * 4 + pcol] = s1;
  }
}

extern "C" void kernel_launch(void* const* d_in, const int* in_sizes, int n_in,
                              void* d_out, int out_size, void* d_ws, size_t ws_size,
                              hipStream_t stream) {
  const float* x    = (const float*)d_in[0];   // (B,1,199) f32
  const float* dirs = (const float*)d_in[1];   // (4,3) f32
  float* out = (float*)d_out;                  // (B,64,4) f32
  const int npts   = in_sizes[0] / D_IN;
  const int blocks = (npts + 7) / 8;           // 8 waves/block, 1 point/wave
  vdx_kernel<<<blocks, 256, 0, stream>>>(x, dirs, out, npts);
}